// GraphLayer_9234179686472
// MI455X (gfx1250) — compile-verified
//
#include <hip/hip_runtime.h>
#include <hip/hip_bf16.h>
#include <math.h>

#define EMBED_DIM 64

typedef float v2f __attribute__((ext_vector_type(2)));
typedef float v8f __attribute__((ext_vector_type(8)));

// ---------------------------------------------------------------------------
// Zero-fill helper (d_out / workspace are poisoned, we must init ourselves).
// ---------------------------------------------------------------------------
__global__ void zero_f32(float* __restrict__ p, int n) {
    int i = blockIdx.x * blockDim.x + threadIdx.x;
    if (i < n) p[i] = 0.0f;
}

// ---------------------------------------------------------------------------
// deg[i] = # edges with src==i   (float accumulator; matches segment_sum of 1s)
// ---------------------------------------------------------------------------
__global__ void degree_kernel(const int* __restrict__ src, float* __restrict__ deg, int E) {
    int e = blockIdx.x * blockDim.x + threadIdx.x;
    if (e < E) atomicAdd(&deg[src[e]], 1.0f);
}

// ---------------------------------------------------------------------------
// h = poi_rep @ W^T + b  via V_WMMA_F32_16X16X4_F32.
// One wave -> one 16x16 output tile. K-loop: 16 WMMA ops (K=64 in steps of 4).
//
// A fragment (16x4 f32, 2 VGPRs): lane half h=lane>>4, m=lane&15
//   a[0] = A[m][k0+2h], a[1] = A[m][k0+2h+1]            (contiguous float2)
// B fragment (4x16 f32, 2 VGPRs): B[k][n] = W[n][k]
//   b[0] = W[n=lane&15][k0+2h], b[1] = W[n][k0+2h+1]    (contiguous float2)
// C/D (16x16 f32, 8 VGPRs): reg r, lane l -> M = r + 8*(l>>4), N = l&15
// ---------------------------------------------------------------------------
__global__ void gemm_wmma_f32(const float* __restrict__ A,   // N x 64 (poi_rep)
                              const float* __restrict__ W,   // 64 x 64 row-major
                              const float* __restrict__ bias, // 64
                              float* __restrict__ H,         // N x 64
                              int numTiles)                  // (N/16)*4
{
    int waveInBlock = threadIdx.x >> 5;
    int lane        = threadIdx.x & 31;
    int tile        = blockIdx.x * (blockDim.x >> 5) + waveInBlock;
    if (tile >= numTiles) return;          // wave-uniform: EXEC stays all-1s

    int rowTile = tile >> 2;               // 4 column tiles of 16 over D=64
    int colTile = tile & 3;
    int row0 = rowTile << 4;
    int col0 = colTile << 4;
    int half = lane >> 4;                  // 0: K pair {0,1}; 1: K pair {2,3}
    int m    = lane & 15;                  // A row index / B col index per lane

    v8f acc = {};
    const float* arow = A + (size_t)(row0 + m) * EMBED_DIM;
    const float* wrow = W + (size_t)(col0 + m) * EMBED_DIM;

#pragma unroll
    for (int k0 = 0; k0 < EMBED_DIM; k0 += 4) {
        int ka = k0 + 2 * half;
        v2f a = *(const v2f*)(arow + ka);
        v2f b = *(const v2f*)(wrow + ka);
        acc = __builtin_amdgcn_wmma_f32_16x16x4_f32(
            /*neg_a=*/false, a, /*neg_b=*/false, b,
            /*c_mod=*/(short)0, acc, /*reuse_a=*/false, /*reuse_b=*/false);
    }

    float bj = bias[col0 + m];
#pragma unroll
    for (int r = 0; r < 8; ++r) {
        int M = row0 + r + 8 * half;
        H[(size_t)M * EMBED_DIM + col0 + m] = acc[r] + bj;
    }
}

// ---------------------------------------------------------------------------
// out[n1] += w * h[n2]  — 64 threads per edge, one f32 atomic per thread.
// h (25.6MB) and out (25.6MB) both live in the 192MB L2 -> L2-resident atomics.
// ---------------------------------------------------------------------------
__global__ void edge_scatter(const float* __restrict__ H,
                             const int*   __restrict__ src,
                             const int*   __restrict__ dst,
                             const float* __restrict__ dist,
                             const float* __restrict__ deg,
                             float* __restrict__ out, int E)
{
    long long t = (long long)blockIdx.x * blockDim.x + threadIdx.x;
    int e = (int)(t >> 6);
    int d = (int)(t & 63);
    if (e >= E) return;
    int n1 = src[e];
    int n2 = dst[e];
    float dv = dist[e];
    float w = rsqrtf(deg[n1] * deg[n2]) * __expf(-dv * dv);
    atomicAdd(&out[(size_t)n1 * EMBED_DIM + d], w * H[(size_t)n2 * EMBED_DIM + d]);
}

// ---------------------------------------------------------------------------
// leaky_relu(0.01) then L2-normalize each row. One wave32 per node, 2 f32/lane.
// ---------------------------------------------------------------------------
__global__ void finalize_kernel(float* __restrict__ out, int N)
{
    int wave = (blockIdx.x * blockDim.x + threadIdx.x) >> 5;
    int lane = threadIdx.x & 31;
    if (wave >= N) return;
    float* row = out + (size_t)wave * EMBED_DIM;
    float2 v = *(float2*)(row + lane * 2);
    v.x = v.x > 0.0f ? v.x : 0.01f * v.x;
    v.y = v.y > 0.0f ? v.y : 0.01f * v.y;
    float ss = v.x * v.x + v.y * v.y;
#pragma unroll
    for (int m = 16; m > 0; m >>= 1) ss += __shfl_xor(ss, m, 32);
    float inv = 1.0f / fmaxf(sqrtf(ss), 1e-12f);
    v.x *= inv; v.y *= inv;
    *(float2*)(row + lane * 2) = v;
}

// ---------------------------------------------------------------------------
// Launcher. Inputs: poi_rep(N*64 f32), edge_index(2*E i32), dist_vec(E f32),
// W(64*64 f32), b(64 f32). Output: N*64 f32.
// ---------------------------------------------------------------------------
extern "C" void kernel_launch(void* const* d_in, const int* in_sizes, int n_in,
                              void* d_out, int out_size, void* d_ws, size_t ws_size,
                              hipStream_t stream)
{
    const float* poi_rep = (const float*)d_in[0];
    const int*   eidx    = (const int*)d_in[1];
    const float* dist    = (const float*)d_in[2];
    const float* W       = (const float*)d_in[3];
    const float* bias    = (const float*)d_in[4];
    float*       out     = (float*)d_out;

    int N = in_sizes[0] / EMBED_DIM;
    int E = in_sizes[1] / 2;
    const int* src = eidx;       // edge_index[0]
    const int* dst = eidx + E;   // edge_index[1]

    // Workspace layout: deg (N f32), then h (N*64 f32), 256B-aligned.
    float* deg = (float*)d_ws;
    size_t degBytes = ((size_t)N * sizeof(float) + 255) & ~(size_t)255;
    float* h = (float*)((char*)d_ws + degBytes);

    const int TB = 256;

    // 1) zero deg and out
    zero_f32<<<(N + TB - 1) / TB, TB, 0, stream>>>(deg, N);
    int outElems = N * EMBED_DIM;
    zero_f32<<<(outElems + TB - 1) / TB, TB, 0, stream>>>(out, outElems);

    // 2) degree histogram
    degree_kernel<<<(E + TB - 1) / TB, TB, 0, stream>>>(src, deg, E);

    // 3) h = poi_rep @ W^T + b   (WMMA f32; 8 waves/block)
    int numTiles = (N / 16) * (EMBED_DIM / 16);           // 6250 * 4 = 25000
    int wavesPerBlock = TB / 32;                          // 8
    int gemmBlocks = (numTiles + wavesPerBlock - 1) / wavesPerBlock;
    gemm_wmma_f32<<<gemmBlocks, TB, 0, stream>>>(poi_rep, W, bias, h, numTiles);

    // 4) edge scatter: 64 threads per edge
    long long scatterThreads = (long long)E * EMBED_DIM;
    int scatterBlocks = (int)((scatterThreads + TB - 1) / TB);
    edge_scatter<<<scatterBlocks, TB, 0, stream>>>(h, src, dst, dist, deg, out, E);

    // 5) leaky-relu + L2 normalize: one wave per node
    int nodesPerBlock = TB / 32;
    finalize_kernel<<<(N + nodesPerBlock - 1) / nodesPerBlock, TB, 0, stream>>>(out, N);
}